// NeuralSort_36687610642842
// MI455X (gfx1250) — compile-verified
//
#include <hip/hip_runtime.h>
#include <math.h>

typedef __attribute__((ext_vector_type(2))) float v2f;
typedef __attribute__((ext_vector_type(8))) float v8f;

#define NN 4096
#define NB 4

// ---------------------------------------------------------------------------
// Kernel 1: B_i = sum_k |s_i - s_k|  via V_WMMA_F32_16X16X4_F32 with B = ones.
// (|s_i - s_k| == |scores_i - scores_k|, negation cancels under abs.)
// One wave handles 16 rows; block = 8 waves = 128 rows. Grid (32, NB).
// A-matrix 16x4 f32 layout (ISA 7.12.2): lanes 0-15 -> K={0,1} in v[0],v[1];
// lanes 16-31 -> K={2,3}. All-ones 4x16 B makes every D column the row sum.
// ---------------------------------------------------------------------------
__global__ void __launch_bounds__(256)
nsort_rowsum_wmma(const float* __restrict__ scores, float* __restrict__ Bout) {
  __shared__ float s_lds[NN];
  const int b = blockIdx.y;
  for (int t = threadIdx.x; t < NN; t += 256)
    s_lds[t] = scores[b * NN + t];
  __syncthreads();

  const int wave = threadIdx.x >> 5;
  const int lane = threadIdx.x & 31;
  const int rowBase = blockIdx.x * 128 + wave * 16;
  const int M = rowBase + (lane & 15);
  const int koff = (lane >> 4) << 1;           // 0 for lanes 0-15, 2 for 16-31
  const float sM = s_lds[M];

  v8f acc = {0.f, 0.f, 0.f, 0.f, 0.f, 0.f, 0.f, 0.f};
  const v2f ones = {1.0f, 1.0f};

  for (int k0 = 0; k0 < NN; k0 += 4) {
    v2f a;
    a.x = fabsf(sM - s_lds[k0 + koff]);
    a.y = fabsf(sM - s_lds[k0 + koff + 1]);
    // D = A x ones(4x16) + C  -> accumulates row sums of the 16x4 tile
    acc = __builtin_amdgcn_wmma_f32_16x16x4_f32(
        false, a, false, ones, (short)0, acc, false, false);
  }

  // C/D layout: VGPR r, lanes 0-15 hold M=r (N=lane); lanes 16-31 hold M=8+r.
  if ((lane & 15) == 0) {
    const int half = lane >> 4;                // 0 -> rows 0-7, 1 -> rows 8-15
#pragma unroll
    for (int r = 0; r < 8; ++r)
      Bout[b * NN + rowBase + half * 8 + r] = acc[r];
  }
}

// ---------------------------------------------------------------------------
// Kernel 2: per-column stats. For each (b, j):
//   m_j = max_i (s_i*scal_j - B_i),  invD_j = 1 / sum_i exp(l - m_j)
// Block = 256 threads = 256 columns, s/B staged in LDS (32 KB). Grid (16, NB).
// ---------------------------------------------------------------------------
__global__ void __launch_bounds__(256)
nsort_colstats(const float* __restrict__ scores, const float* __restrict__ B,
               float* __restrict__ m_out, float* __restrict__ invd_out) {
  __shared__ float s_s[NN];
  __shared__ float s_B[NN];
  const int b = blockIdx.y;
  for (int t = threadIdx.x; t < NN; t += 256) {
    s_s[t] = -scores[b * NN + t];              // s = -scores
    s_B[t] = B[b * NN + t];
  }
  __syncthreads();

  const int j = blockIdx.x * 256 + threadIdx.x;
  const float scal = (float)(NN - 1 - 2 * j);

  float m = -INFINITY;
  for (int i = 0; i < NN; ++i)
    m = fmaxf(m, fmaf(s_s[i], scal, -s_B[i]));

  float d = 0.f;
  for (int i = 0; i < NN; ++i)
    d += __expf(fmaf(s_s[i], scal, -s_B[i]) - m);

  m_out[b * NN + j] = m;
  invd_out[b * NN + j] = 1.0f / d;
}

// ---------------------------------------------------------------------------
// Kernel 3: streaming store pass (the 256 MB, bandwidth-bound part).
// out[b,i,j] = exp(s_i*scal_j - B_i - m_j) * invD_j, float4 along contiguous j.
// ---------------------------------------------------------------------------
__global__ void __launch_bounds__(256)
nsort_emit(const float* __restrict__ scores, const float* __restrict__ B,
           const float* __restrict__ m, const float* __restrict__ invd,
           float4* __restrict__ out) {
  const int JV = NN / 4;
  const int idx = blockIdx.x * 256 + threadIdx.x;   // < 16,777,216
  const int jv = idx & (JV - 1);
  const int rest = idx >> 10;                       // idx / JV
  const int i = rest & (NN - 1);
  const int b = rest >> 12;                         // rest / NN

  const float si = -scores[b * NN + i];
  const float Bi = B[b * NN + i];
  const float4 mv = ((const float4*)(m + b * NN))[jv];
  const float4 dv = ((const float4*)(invd + b * NN))[jv];
  const int j0 = jv * 4;

  float4 r;
  r.x = __expf(fmaf(si, (float)(NN - 1 - 2 * (j0 + 0)), -Bi) - mv.x) * dv.x;
  r.y = __expf(fmaf(si, (float)(NN - 1 - 2 * (j0 + 1)), -Bi) - mv.y) * dv.y;
  r.z = __expf(fmaf(si, (float)(NN - 1 - 2 * (j0 + 2)), -Bi) - mv.z) * dv.z;
  r.w = __expf(fmaf(si, (float)(NN - 1 - 2 * (j0 + 3)), -Bi) - mv.w) * dv.w;
  out[idx] = r;
}

// ---------------------------------------------------------------------------
extern "C" void kernel_launch(void* const* d_in, const int* in_sizes, int n_in,
                              void* d_out, int out_size, void* d_ws, size_t ws_size,
                              hipStream_t stream) {
  (void)in_sizes; (void)n_in; (void)out_size; (void)ws_size;
  const float* scores = (const float*)d_in[0];

  // Workspace layout: B | m | invD, each NB*NN floats (64 KB each).
  float* Bws   = (float*)d_ws;
  float* mws   = Bws + NB * NN;
  float* dws   = mws + NB * NN;

  nsort_rowsum_wmma<<<dim3(32, NB), 256, 0, stream>>>(scores, Bws);
  nsort_colstats<<<dim3(16, NB), 256, 0, stream>>>(scores, Bws, mws, dws);

  const int total_vec4 = NB * NN * (NN / 4);        // 16,777,216
  nsort_emit<<<total_vec4 / 256, 256, 0, stream>>>(scores, Bws, mws, dws,
                                                   (float4*)d_out);
}